// SAGE_6691559047385
// MI455X (gfx1250) — compile-verified
//
#include <hip/hip_runtime.h>

// GraphSAGE (2x SAGEConv, mean aggregation) for MI455X / gfx1250.
//
// Cost model: edge scatter (~5 GB, L2-resident: x=51MB, h=102MB < 192MB L2)
// dominates; GEMMs are skinny/memory-bound (~18 GFLOP) so we keep full f32
// precision and use V_WMMA_F32_16X16X4_F32 (wave32, one 16x16 tile per wave).
// The mean (agg/deg) is fused into the GEMM A-operand to delete two full
// read+write passes over the aggregate buffers.

typedef __attribute__((ext_vector_type(2))) float v2f;
typedef __attribute__((ext_vector_type(8))) float v8f;

#define N_FEAT   128
#define HIDN     256
#define N_CLS    47

// ---------------------------------------------------------------- zero ws ---
__global__ void sage_zero_kernel(float4* __restrict__ p, long long n4) {
  long long i = (long long)blockIdx.x * blockDim.x + threadIdx.x;
  if (i < n4) p[i] = make_float4(0.f, 0.f, 0.f, 0.f);
}

// ------------------------------------------------------------- scatter-add --
// One thread per (edge, 4 consecutive floats): a wave32 covers 128 floats of
// one edge's row -> coalesced 512B gather per edge, atomics land in L2.
// logPer = log2(F/4) so edge/feature split is shifts, not 64-bit division.
__global__ void sage_scatter_kernel(const float* __restrict__ feat,
                                    const int* __restrict__ src,
                                    const int* __restrict__ dst,
                                    float* __restrict__ agg,
                                    float* __restrict__ deg,
                                    long long nE, int F, int logPer, int addDeg) {
  long long tid = (long long)blockIdx.x * blockDim.x + threadIdx.x;
  long long e = tid >> logPer;
  if (e >= nE) return;
  int f4 = ((int)tid & ((1 << logPer) - 1)) << 2;
  int s = src[e];
  int d = dst[e];
  const float4 v = *(const float4*)(feat + (size_t)s * F + f4);
  float* ap = agg + (size_t)d * F + f4;
  atomicAdd(ap + 0, v.x);
  atomicAdd(ap + 1, v.y);
  atomicAdd(ap + 2, v.z);
  atomicAdd(ap + 3, v.w);
  if (addDeg && f4 == 0) atomicAdd(deg + d, 1.0f);
}

// ------------------------------------------------ fused SAGE GEMM (WMMA) ----
// out[M x Nout] = (A0/deg)[M x K0] @ B0 + A1[M x K1] @ B1 + bias  (opt. ReLU)
// One wave per 16x16 output tile; K stepped by 4 via V_WMMA_F32_16X16X4_F32.
//
// Fragment layout (ISA 7.12.2, 16x4 f32 A / 4x16 f32 B / 16x16 f32 C):
//   lane = 16*half + lidx
//   A: v0 = A[row=lidx][k0+2*half],  v1 = A[row=lidx][k0+2*half+1]
//   B: v0 = B[k0+2*half][col=lidx],  v1 = B[k0+2*half+1][col=lidx]
//   C: acc[r] = C[row = 8*half + r][col = lidx]
//
// Out-of-range output columns (Nout=47 padded to 48) are handled by CLAMPING
// the B column index: the padded column computes garbage but is never stored,
// so the inner loop has no divergent predication (no saveexec churn).
__global__ void sage_gemm_wmma_kernel(const float* __restrict__ A0,
                                      const float* __restrict__ A1,
                                      const float* __restrict__ B0,
                                      const float* __restrict__ B1,
                                      const float* __restrict__ bias,
                                      const float* __restrict__ deg,
                                      float* __restrict__ out,
                                      int K0, int K1, int Ntile, int Nout,
                                      int nRowTiles, int M, int doRelu) {
  int gtid = blockIdx.x * blockDim.x + threadIdx.x;
  int w    = gtid >> 5;          // one 16x16 tile per wave
  int lane = gtid & 31;
  int colTiles = Ntile >> 4;
  int rt = w / colTiles;
  int ct = w - rt * colTiles;
  if (rt >= nRowTiles) return;   // wave-uniform: EXEC stays all-ones for WMMA

  int half = lane >> 4;          // 0: lanes 0-15, 1: lanes 16-31
  int lidx = lane & 15;
  int rowBase = rt << 4;
  int arow = rowBase + lidx;
  if (arow >= M) arow = M - 1;   // safety clamp (M=100000 is /16 exact)
  int col = (ct << 4) + lidx;
  bool colOK = (col < Nout);
  int bcol = colOK ? col : (Nout - 1);   // clamped load column

  // fused mean: A0 rows are sums; scale by 1/max(deg,1) on the fly
  float inv = 1.0f / fmaxf(deg[arow], 1.0f);

  v8f acc = {};

  // ---- (A0/deg) @ B0 : aggregated-neighbor branch ----
  {
    const float* ap = A0 + (size_t)arow * K0 + 2 * half;
    const float* bp = B0 + (size_t)(2 * half) * Nout + bcol;
    const size_t bstep = (size_t)4 * Nout;
#pragma unroll 8
    for (int k = 0; k < K0; k += 4) {
      v2f a;
      a.x = ap[0] * inv;
      a.y = ap[1] * inv;
      v2f b;
      b.x = bp[0];
      b.y = bp[Nout];
      acc = __builtin_amdgcn_wmma_f32_16x16x4_f32(
          /*neg_a=*/false, a, /*neg_b=*/false, b,
          /*c_mod=*/(short)0, acc, /*reuse_a=*/false, /*reuse_b=*/false);
      ap += 4;
      bp += bstep;
    }
  }

  // ---- A1 @ B1 : self/root branch ----
  {
    const float* ap = A1 + (size_t)arow * K1 + 2 * half;
    const float* bp = B1 + (size_t)(2 * half) * Nout + bcol;
    const size_t bstep = (size_t)4 * Nout;
#pragma unroll 8
    for (int k = 0; k < K1; k += 4) {
      v2f a;
      a.x = ap[0];
      a.y = ap[1];
      v2f b;
      b.x = bp[0];
      b.y = bp[Nout];
      acc = __builtin_amdgcn_wmma_f32_16x16x4_f32(
          false, a, false, b, (short)0, acc, false, false);
      ap += 4;
      bp += bstep;
    }
  }

  // ---- epilogue: bias (+ReLU) + guarded store ----
  if (colOK) {
    float bv = bias[col];
    int orow0 = rowBase + (half << 3);
    float* op = out + (size_t)orow0 * Nout + col;
#pragma unroll
    for (int r = 0; r < 8; ++r) {
      if (orow0 + r < M) {
        float vv = acc[r] + bv;
        if (doRelu) vv = fmaxf(vv, 0.0f);
        *op = vv;
      }
      op += Nout;
    }
  }
}

// ------------------------------------------------------------------ driver --
extern "C" void kernel_launch(void* const* d_in, const int* in_sizes, int n_in,
                              void* d_out, int out_size, void* d_ws, size_t ws_size,
                              hipStream_t stream) {
  const float* x   = (const float*)d_in[0];   // [N, 128]
  const int*   ei  = (const int*)d_in[1];     // [2, E] row-major
  const float* Wl1 = (const float*)d_in[2];   // [128, 256]
  const float* Wr1 = (const float*)d_in[3];   // [128, 256]
  const float* b1  = (const float*)d_in[4];   // [256]
  const float* Wl2 = (const float*)d_in[5];   // [256, 47]
  const float* Wr2 = (const float*)d_in[6];   // [256, 47]
  const float* b2  = (const float*)d_in[7];   // [47]

  const int N = in_sizes[0] / N_FEAT;              // 100000
  const long long E = (long long)in_sizes[1] / 2;  // 1600000
  const int* src = ei;
  const int* dst = ei + E;

  // Workspace layout (contiguous so the zero-fill is one range):
  //   deg [N] | agg1 [N*128] | agg2 [N*256] | h [N*256]   -> ~256.4 MB
  float* deg  = (float*)d_ws;
  float* agg1 = deg + N;
  float* agg2 = agg1 + (size_t)N * N_FEAT;
  float* h    = agg2 + (size_t)N * HIDN;

  const int BLK = 256;

  // 0) zero deg+agg1+agg2 (N*(1+128+256) floats; N%4==0 so float4 is exact)
  long long n4 = (long long)N * (1 + N_FEAT + HIDN) / 4;
  sage_zero_kernel<<<(unsigned)((n4 + BLK - 1) / BLK), BLK, 0, stream>>>(
      (float4*)d_ws, n4);

  // 1) scatter x along edges into agg1, count degrees
  long long t1 = E * (N_FEAT / 4);
  sage_scatter_kernel<<<(unsigned)((t1 + BLK - 1) / BLK), BLK, 0, stream>>>(
      x, src, dst, agg1, deg, E, N_FEAT, 5, 1);

  // 2) h = relu((agg1/deg) @ Wl1 + x @ Wr1 + b1)   [K=128+128, Nout=256]
  int rowTiles = (N + 15) / 16;
  int waves1 = rowTiles * (HIDN / 16);
  sage_gemm_wmma_kernel<<<(unsigned)((waves1 + 7) / 8), BLK, 0, stream>>>(
      agg1, x, Wl1, Wr1, b1, deg, h, N_FEAT, N_FEAT, HIDN, HIDN, rowTiles, N, 1);

  // 3) scatter h along edges into agg2
  long long t2 = E * (HIDN / 4);
  sage_scatter_kernel<<<(unsigned)((t2 + BLK - 1) / BLK), BLK, 0, stream>>>(
      h, src, dst, agg2, nullptr, E, HIDN, 6, 0);

  // 4) out = (agg2/deg) @ Wl2 + h @ Wr2 + b2   [K=256+256, Nout=47 pad 48]
  int waves2 = rowTiles * (48 / 16);
  sage_gemm_wmma_kernel<<<(unsigned)((waves2 + 7) / 8), BLK, 0, stream>>>(
      agg2, h, Wl2, Wr2, b2, deg, (float*)d_out, HIDN, HIDN, 48, N_CLS,
      rowTiles, N, 0);
}